// GNN_89953795047502
// MI455X (gfx1250) — compile-verified
//
#include <hip/hip_runtime.h>
#include <hip/hip_bf16.h>

#define NN   50000
#define EE   800000
#define FIN  32
#define HH   128
#define EDIM 21
#define PP   64
#define ETOT (EE + NN)
#define NEG_SLOPE 0.2f

typedef float v2f __attribute__((ext_vector_type(2)));
typedef float v8f __attribute__((ext_vector_type(8)));
typedef unsigned u32x4 __attribute__((ext_vector_type(4)));
typedef int i32x4 __attribute__((ext_vector_type(4)));
typedef int i32x8 __attribute__((ext_vector_type(8)));

#if __has_builtin(__builtin_amdgcn_tensor_load_to_lds) && \
    __has_builtin(__builtin_amdgcn_s_wait_tensorcnt)
#define HAVE_TDM 1
#else
#define HAVE_TDM 0
#endif

// ---- ordered-uint mapping for float atomic max (handles negatives) ----
__device__ __forceinline__ unsigned ordf(float f) {
  unsigned u = __float_as_uint(f);
  return (u & 0x80000000u) ? ~u : (u | 0x80000000u);
}
__device__ __forceinline__ float unordf(unsigned u) {
  return (u & 0x80000000u) ? __uint_as_float(u ^ 0x80000000u)
                           : __uint_as_float(~u);
}

// ============================ fill kernels ============================
__global__ void fill_f32(float* p, float v, int n) {
  for (int i = blockIdx.x * blockDim.x + threadIdx.x; i < n;
       i += gridDim.x * blockDim.x)
    p[i] = v;
}
__global__ void fill_u32(unsigned* p, unsigned v, int n) {
  for (int i = blockIdx.x * blockDim.x + threadIdx.x; i < n;
       i += gridDim.x * blockDim.x)
    p[i] = v;
}
__global__ void scale_f32(float* p, float s, int n) {
  for (int i = blockIdx.x * blockDim.x + threadIdx.x; i < n;
       i += gridDim.x * blockDim.x)
    p[i] *= s;
}

// ===================== fp32 WMMA GEMM: C = A@W + b =====================
// A:[M,K] row-major, W:[K,128] row-major, C:[M,128]. M % 16 == 0.
// Block = 256 threads = 8 waves; block tile = 16 rows x 128 cols,
// each wave owns one 16x16 N-tile, loops K in steps of 4 with
// V_WMMA_F32_16X16X4_F32 (fp32 A/B, fp32 accumulate).
// A tile staged into LDS as [16][K+1] (padded): via the Tensor Data Mover
// when available (pad_enable inserts 1 DWORD after every K DWORDs), else
// via a cooperative vector load.
template <int K, bool RELU>
__global__ __launch_bounds__(256) void gemm_wmma(
    const float* __restrict__ A, const float* __restrict__ W,
    const float* __restrict__ bias, float* __restrict__ C) {
  constexpr int KP = K + 1;  // padded row to break LDS bank conflicts
  __shared__ float As[16 * KP];
  const int m0 = blockIdx.x * 16;
  const int tid = threadIdx.x;

  // warm L0/L2 with the (L2-resident) weight matrix
  for (int off = tid * 256; off < K * HH * 4; off += 256 * 256)
    __builtin_prefetch((const char*)W + off, 0, 3);

#if HAVE_TDM
  if (tid < 32) {  // one wave issues the TDM transfer (per-wave instruction)
    constexpr unsigned PADCODE = (K == 32) ? 4u : (K == 128) ? 6u : 7u;
    const unsigned long long ga =
        (unsigned long long)(const void*)(A + (size_t)m0 * K);
    u32x4 g0;
    g0[0] = 1u;                                    // count=1 (valid), no gather
    g0[1] = 0u;                                    // lds_addr = 0 (As at base)
    g0[2] = (unsigned)(ga & 0xFFFFFFFFu);          // global_addr[31:0]
    g0[3] = (unsigned)((ga >> 32) & 0x01FFFFFFu)   // global_addr[56:32]
            | (2u << 30);                          // type = 2 ("image")
    i32x8 g1;
    g1[0] = (int)((2u << 16)                       // data_size = 4 bytes
                  | (1u << 20)                     // pad_enable
                  | (PADCODE << 22));              // pad_interval = K dwords
                                                   // pad_amount = 0 (1 dword)
    g1[1] = (int)(((unsigned)K & 0xFFFFu) << 16);  // tensor_dim0[15:0]
    g1[2] = (int)((((unsigned)K >> 16) & 0xFFFFu)  // tensor_dim0[31:16]
                  | (16u << 16));                  // tensor_dim1[15:0] = 16
    g1[3] = (int)(((unsigned)K) << 16);            // tile_dim0 = K
    g1[4] = (int)16u;                              // tile_dim1 = 16, tile_dim2=0
    g1[5] = (int)K;                                // tensor_dim0_stride[31:0]
    g1[6] = 0;                                     // stride hi / dim1_stride lo
    g1[7] = 0;
    i32x4 gz4 = {0, 0, 0, 0};
    i32x8 gz8 = {0, 0, 0, 0, 0, 0, 0, 0};
    __builtin_amdgcn_tensor_load_to_lds(g0, g1, gz4, gz4, gz8, 0);
    __builtin_amdgcn_s_wait_tensorcnt(0);
  }
  __syncthreads();
#else
  for (int i = tid; i < 16 * K; i += 256) {
    int r = i / K, c = i - r * K;
    As[r * KP + c] = A[(size_t)(m0 + r) * K + c];
  }
  __syncthreads();
#endif

  const int wv = tid >> 5;       // wave id -> N tile
  const int lane = tid & 31;
  const int n0 = wv * 16;
  const int hi = lane >> 4;      // 0: lanes 0-15, 1: lanes 16-31
  const int l = lane & 15;

  v8f acc = {};
  for (int k0 = 0; k0 < K; k0 += 4) {
    const int ka = k0 + hi * 2;  // A/B VGPR0 holds {K0,K2}, VGPR1 {K1,K3}
    v2f a, b;
    a.x = As[l * KP + ka];
    a.y = As[l * KP + ka + 1];
    b.x = W[(size_t)ka * HH + n0 + l];
    b.y = W[(size_t)(ka + 1) * HH + n0 + l];
    acc = __builtin_amdgcn_wmma_f32_16x16x4_f32(false, a, false, b, (short)0,
                                                acc, false, false);
  }
  const float bc = bias[n0 + l];
#pragma unroll
  for (int r = 0; r < 8; ++r) {  // VGPR r -> rows {r, r+8}
    float v = acc[r] + bc;
    if (RELU) v = fmaxf(v, 0.0f);
    C[(size_t)(m0 + r + hi * 8) * HH + n0 + l] = v;
  }
}

// ======================= edge-attr mean helpers =======================
__global__ __launch_bounds__(256) void edge_attr_sum(
    const float* __restrict__ eattr, float* accum) {
  float loc[EDIM];
#pragma unroll
  for (int k = 0; k < EDIM; ++k) loc[k] = 0.0f;
  for (int e = blockIdx.x * blockDim.x + threadIdx.x; e < EE;
       e += gridDim.x * blockDim.x) {
    const float* p = eattr + (size_t)e * EDIM;
#pragma unroll
    for (int k = 0; k < EDIM; ++k) loc[k] += p[k];
  }
#pragma unroll
  for (int k = 0; k < EDIM; ++k) atomicAdd(&accum[k], loc[k]);
}

__global__ void ewm_kernel(const float* __restrict__ emean,
                           const float* __restrict__ We, float* ewm) {
  int j = threadIdx.x;  // 128 threads
  float s = 0.0f;
#pragma unroll
  for (int k = 0; k < EDIM; ++k) s += emean[k] * We[k * HH + j];
  ewm[j] = s;
}

// ============== pass 1: alpha = att . leaky(xl[s]+xr[d]+eW) ==============
__global__ __launch_bounds__(256) void edge_alpha(
    const int* __restrict__ src, const int* __restrict__ dst,
    const float* __restrict__ eattr, const float* __restrict__ We,
    const float* __restrict__ att, const float* __restrict__ ewm,
    const float* __restrict__ xl, const float* __restrict__ xr,
    float* __restrict__ alpha, unsigned* __restrict__ amax) {
  __shared__ float sWe[EDIM * HH];
  __shared__ float sAtt[HH];
  const int tid = threadIdx.x;
  const int wv = tid >> 5, lane = tid & 31;
  const int e = blockIdx.x * 8 + wv;

  // prefetch this wave's edge-attr row while we stage We into LDS
  if (e < EE) __builtin_prefetch((const char*)(eattr + (size_t)e * EDIM), 0, 0);

  for (int i = tid; i < EDIM * HH; i += 256) sWe[i] = We[i];
  for (int i = tid; i < HH; i += 256) sAtt[i] = att[i];
  __syncthreads();

  if (e >= ETOT) return;

  const bool self = (e >= EE);
  const int s = self ? (e - EE) : src[e];
  const int d = self ? s : dst[e];
  const int f = lane * 4;

  float4 vl = *(const float4*)(xl + (size_t)s * HH + f);
  float4 vr = *(const float4*)(xr + (size_t)d * HH + f);
  float a0, a1, a2, a3;
  if (!self) {
    const float* ep = eattr + (size_t)e * EDIM;
    a0 = a1 = a2 = a3 = 0.0f;
#pragma unroll
    for (int k = 0; k < EDIM; ++k) {
      const float ek = ep[k];
      const float* wr = sWe + k * HH + f;
      a0 += ek * wr[0]; a1 += ek * wr[1]; a2 += ek * wr[2]; a3 += ek * wr[3];
    }
  } else {
    float4 em = *(const float4*)(ewm + f);
    a0 = em.x; a1 = em.y; a2 = em.z; a3 = em.w;
  }
  float m0 = vl.x + vr.x + a0, m1 = vl.y + vr.y + a1;
  float m2 = vl.z + vr.z + a2, m3 = vl.w + vr.w + a3;
  m0 = m0 > 0.f ? m0 : NEG_SLOPE * m0;
  m1 = m1 > 0.f ? m1 : NEG_SLOPE * m1;
  m2 = m2 > 0.f ? m2 : NEG_SLOPE * m2;
  m3 = m3 > 0.f ? m3 : NEG_SLOPE * m3;

  float p = m0 * sAtt[f] + m1 * sAtt[f + 1] + m2 * sAtt[f + 2] + m3 * sAtt[f + 3];
#pragma unroll
  for (int off = 16; off > 0; off >>= 1) p += __shfl_xor(p, off, 32);
  if (lane == 0) {
    alpha[e] = p;
    atomicMax(&amax[d], ordf(p));
  }
}

// =========== pass 2: ex = exp(alpha - amax[d]); denom[d] += ex ===========
__global__ __launch_bounds__(256) void edge_exp(const int* __restrict__ dst,
                                                float* __restrict__ alpha,
                                                const unsigned* __restrict__ amax,
                                                float* __restrict__ denom) {
  for (int e = blockIdx.x * blockDim.x + threadIdx.x; e < ETOT;
       e += gridDim.x * blockDim.x) {
    const int d = (e >= EE) ? (e - EE) : dst[e];
    const float ex = __expf(alpha[e] - unordf(amax[d]));
    alpha[e] = ex;
    atomicAdd(&denom[d], ex);
  }
}

// ============ pass 3: out[d] += (ex/denom[d]) * xl[s] ============
__global__ __launch_bounds__(256) void edge_aggregate(
    const int* __restrict__ src, const int* __restrict__ dst,
    const float* __restrict__ ex, const float* __restrict__ denom,
    const float* __restrict__ xl, float* __restrict__ out) {
  const int tid = threadIdx.x;
  const int wv = tid >> 5, lane = tid & 31;
  const int e = blockIdx.x * 8 + wv;
  if (e >= ETOT) return;
  const bool self = (e >= EE);
  const int s = self ? (e - EE) : src[e];
  const int d = self ? s : dst[e];
  const float w = ex[e] / denom[d];
  const int f = lane * 4;
  float4 v = *(const float4*)(xl + (size_t)s * HH + f);
  float* o = out + (size_t)d * HH + f;
  atomicAdd(o + 0, v.x * w);
  atomicAdd(o + 1, v.y * w);
  atomicAdd(o + 2, v.z * w);
  atomicAdd(o + 3, v.w * w);
}

__global__ void bias_relu(float* h, const float* __restrict__ bias, int total) {
  for (int i = blockIdx.x * blockDim.x + threadIdx.x; i < total;
       i += gridDim.x * blockDim.x)
    h[i] = fmaxf(h[i] + bias[i & (HH - 1)], 0.0f);
}

// ========================= segment mean pooling =========================
__global__ void pool_acc(const float* __restrict__ h, const int* __restrict__ seg,
                         float* means) {
  for (int i = blockIdx.x * blockDim.x + threadIdx.x; i < NN * HH;
       i += gridDim.x * blockDim.x) {
    const int n = i >> 7, f = i & (HH - 1);
    atomicAdd(&means[seg[n] * HH + f], h[i]);
  }
}
__global__ void cnt_acc(const int* __restrict__ seg, float* cnt) {
  for (int n = blockIdx.x * blockDim.x + threadIdx.x; n < NN;
       n += gridDim.x * blockDim.x)
    atomicAdd(&cnt[seg[n]], 1.0f);
}
__global__ void pool_div(float* means, const float* __restrict__ cnt) {
  for (int i = blockIdx.x * blockDim.x + threadIdx.x; i < PP * HH;
       i += gridDim.x * blockDim.x)
    means[i] /= cnt[i >> 7];
}
__global__ void build_u(const float* __restrict__ h, const float* __restrict__ means,
                        const int* __restrict__ seg, float* __restrict__ u) {
  for (int i = blockIdx.x * blockDim.x + threadIdx.x; i < NN * HH;
       i += gridDim.x * blockDim.x) {
    const int n = i >> 7, f = i & (HH - 1);
    u[(size_t)n * (2 * HH) + f] = h[i];
    u[(size_t)n * (2 * HH) + HH + f] = means[seg[n] * HH + f];
  }
}

// ================== final [N,128] @ [128,1] + b ==================
__global__ __launch_bounds__(256) void dot_out(const float* __restrict__ h,
                                               const float* __restrict__ w,
                                               const float* __restrict__ b,
                                               float* __restrict__ out) {
  const int tid = threadIdx.x;
  const int wv = tid >> 5, lane = tid & 31;
  const int row = blockIdx.x * 8 + wv;
  if (row >= NN) return;
  const int f = lane * 4;
  float4 v = *(const float4*)(h + (size_t)row * HH + f);
  float4 ww = *(const float4*)(w + f);
  float p = v.x * ww.x + v.y * ww.y + v.z * ww.z + v.w * ww.w;
#pragma unroll
  for (int off = 16; off > 0; off >>= 1) p += __shfl_xor(p, off, 32);
  if (lane == 0) out[row] = p + b[0];
}

// ============================ launch ============================
extern "C" void kernel_launch(void* const* d_in, const int* in_sizes, int n_in,
                              void* d_out, int out_size, void* d_ws, size_t ws_size,
                              hipStream_t stream) {
  const float* G     = (const float*)d_in[0];
  const int*   ei    = (const int*)d_in[1];
  const float* eattr = (const float*)d_in[2];
  const int*   seg   = (const int*)d_in[3];
  int ix = 4;
  const float* W_emb1 = (const float*)d_in[ix++]; const float* b_emb1 = (const float*)d_in[ix++];
  const float* W_emb2 = (const float*)d_in[ix++]; const float* b_emb2 = (const float*)d_in[ix++];
  struct Conv { const float *Wl, *bl, *Wr, *br, *We, *att, *bias; } cv[2];
  for (int c = 0; c < 2; ++c) {  // gat() insertion order: Wl,bl,Wr,br,We,att,bias
    cv[c].Wl = (const float*)d_in[ix++]; cv[c].bl = (const float*)d_in[ix++];
    cv[c].Wr = (const float*)d_in[ix++]; cv[c].br = (const float*)d_in[ix++];
    cv[c].We = (const float*)d_in[ix++]; cv[c].att = (const float*)d_in[ix++];
    cv[c].bias = (const float*)d_in[ix++];
  }
  const float* W_g1  = (const float*)d_in[ix++]; const float* b_g1  = (const float*)d_in[ix++];
  const float* W_g2  = (const float*)d_in[ix++]; const float* b_g2  = (const float*)d_in[ix++];
  const float* W_lin = (const float*)d_in[ix++]; const float* b_lin = (const float*)d_in[ix++];
  const float* W_lin2= (const float*)d_in[ix++]; const float* b_lin2= (const float*)d_in[ix++];
  const float* W_lin3= (const float*)d_in[ix++]; const float* b_lin3= (const float*)d_in[ix++];

  const int* src = ei;
  const int* dst = ei + EE;

  // workspace carve (all 16B-aligned)
  float* buf0  = (float*)d_ws;
  float* buf1  = buf0 + (size_t)NN * HH;
  float* xl    = buf1 + (size_t)NN * HH;
  float* xr    = xl + (size_t)NN * HH;
  float* ubuf  = xr + (size_t)NN * HH;
  float* alpha = ubuf + (size_t)NN * 2 * HH;
  float* denom = alpha + ETOT;
  float* means = denom + NN;
  float* cnt   = means + PP * HH;
  float* emean = cnt + PP;
  float* ewm   = emean + 32;
  unsigned* amax = (unsigned*)(ewm + HH);

  const int gM = NN / 16;            // 3125, exact
  const int gE = (ETOT + 7) / 8;     // 8 edges (waves) per block

  // embeddings
  gemm_wmma<FIN, true><<<gM, 256, 0, stream>>>(G, W_emb1, b_emb1, buf0);
  gemm_wmma<HH, true><<<gM, 256, 0, stream>>>(buf0, W_emb2, b_emb2, buf1);

  // mean edge attribute (for self-loop fill_value='mean')
  fill_f32<<<1, 32, 0, stream>>>(emean, 0.0f, EDIM);
  edge_attr_sum<<<64, 256, 0, stream>>>(eattr, emean);
  scale_f32<<<1, 32, 0, stream>>>(emean, 1.0f / (float)EE, EDIM);

  float* hcur = buf1;
  float* hout = buf0;
  for (int c = 0; c < 2; ++c) {
    ewm_kernel<<<1, HH, 0, stream>>>(emean, cv[c].We, ewm);
    gemm_wmma<HH, false><<<gM, 256, 0, stream>>>(hcur, cv[c].Wl, cv[c].bl, xl);
    gemm_wmma<HH, false><<<gM, 256, 0, stream>>>(hcur, cv[c].Wr, cv[c].br, xr);
    fill_u32<<<256, 256, 0, stream>>>(amax, 0u, NN);
    fill_f32<<<256, 256, 0, stream>>>(denom, 0.0f, NN);
    fill_f32<<<1024, 256, 0, stream>>>(hout, 0.0f, NN * HH);
    edge_alpha<<<gE, 256, 0, stream>>>(src, dst, eattr, cv[c].We, cv[c].att,
                                       ewm, xl, xr, alpha, amax);
    edge_exp<<<2048, 256, 0, stream>>>(dst, alpha, amax, denom);
    edge_aggregate<<<gE, 256, 0, stream>>>(src, dst, alpha, denom, xl, hout);
    bias_relu<<<1024, 256, 0, stream>>>(hout, cv[c].bias, NN * HH);
    float* t = hcur; hcur = hout; hout = t;
  }
  // hcur = buf1 (conv2 output), buf0 free
  gemm_wmma<HH, true><<<gM, 256, 0, stream>>>(buf1, W_g1, b_g1, buf0);
  gemm_wmma<HH, false><<<gM, 256, 0, stream>>>(buf0, W_g2, b_g2, buf1);  // final h

  // segment mean pooling + concat
  fill_f32<<<16, 256, 0, stream>>>(means, 0.0f, PP * HH);
  fill_f32<<<1, 64, 0, stream>>>(cnt, 0.0f, PP);
  pool_acc<<<1024, 256, 0, stream>>>(buf1, seg, means);
  cnt_acc<<<256, 256, 0, stream>>>(seg, cnt);
  pool_div<<<32, 256, 0, stream>>>(means, cnt);
  build_u<<<1024, 256, 0, stream>>>(buf1, means, seg, ubuf);

  // head
  gemm_wmma<2 * HH, true><<<gM, 256, 0, stream>>>(ubuf, W_lin, b_lin, buf0);
  gemm_wmma<HH, true><<<gM, 256, 0, stream>>>(buf0, W_lin2, b_lin2, xl);
  dot_out<<<(NN + 7) / 8, 256, 0, stream>>>(xl, W_lin3, b_lin3, (float*)d_out);
}